// CalibrationNetwork_8873402433722
// MI455X (gfx1250) — compile-verified
//
#include <hip/hip_runtime.h>
#include <math.h>

typedef __attribute__((ext_vector_type(2))) float v2f;
typedef __attribute__((ext_vector_type(8))) float v8f;

#define B_N 32768
#define J_N 32
#define Q_N 7
#define O_N 5
#define H1N 64
#define H2N 64
#define KP1 8     // padded (O+1) -> 8
#define KP2 68    // padded (H1+1) -> 68 (17 k-steps of 4)
#define WS2 72    // LDS row stride for staged Wc2^T (bank-conflict-free)
#define ZS  68    // LDS row stride for activation tiles

// ---------- weight combination: Wc1^T[j][q][d][h] = W1[q][h][d] + W1_a[j][q][h][d]
__global__ void combine_w1(const float* __restrict__ W1, const float* __restrict__ W1a,
                           float* __restrict__ wc1) {
    int t = blockIdx.x * blockDim.x + threadIdx.x;
    if (t >= J_N * Q_N * KP1 * H1N) return;
    int h = t & 63;
    int d = (t >> 6) & 7;
    int q = (t >> 9) % Q_N;
    int j = t / (Q_N * KP1 * H1N);
    float v = 0.f;
    if (d < O_N + 1)
        v = W1[(q * H1N + h) * (O_N + 1) + d] +
            W1a[((j * Q_N + q) * H1N + h) * (O_N + 1) + d];
    wc1[t] = v;
}

// ---------- Wc2^T[j][i][h] = W2[h][i] + W2_a[j][h][i], i padded to 68
__global__ void combine_w2(const float* __restrict__ W2, const float* __restrict__ W2a,
                           float* __restrict__ wc2) {
    int t = blockIdx.x * blockDim.x + threadIdx.x;
    if (t >= J_N * KP2 * H2N) return;
    int h = t & 63;
    int i = (t >> 6) % KP2;
    int j = t / (KP2 * H2N);
    float v = 0.f;
    if (i < H1N + 1)
        v = W2[h * (H1N + 1) + i] + W2a[(j * H2N + h) * (H1N + 1) + i];
    wc2[t] = v;
}

// ---------- Vc[j][q][o][i] = V[q][o][i] + V_a[j][q][o][i]
__global__ void combine_vc(const float* __restrict__ V, const float* __restrict__ Va,
                           float* __restrict__ vc) {
    int t = blockIdx.x * blockDim.x + threadIdx.x;
    if (t >= J_N * Q_N * O_N * (H2N + 1)) return;
    int i = t % (H2N + 1);
    int o = (t / (H2N + 1)) % O_N;
    int q = (t / ((H2N + 1) * O_N)) % Q_N;
    vc[t] = V[(q * O_N + o) * (H2N + 1) + i] + Va[t];
}

// ---------- judge binning
__global__ void count_kernel(const int* __restrict__ jid, int* __restrict__ cnt) {
    int b = blockIdx.x * blockDim.x + threadIdx.x;
    if (b < B_N) atomicAdd(&cnt[jid[b]], 1);
}

__global__ void scan_kernel(const int* __restrict__ cnt, int* __restrict__ offs) {
    if (threadIdx.x == 0) {
        int acc = 0;
        offs[0] = 0;
        for (int j = 0; j < J_N; ++j) { acc += cnt[j]; offs[j + 1] = acc; }
    }
}

__global__ void scatter_kernel(const int* __restrict__ jid, const int* __restrict__ offs,
                               int* __restrict__ cur, int* __restrict__ list) {
    int b = blockIdx.x * blockDim.x + threadIdx.x;
    if (b < B_N) {
        int j = jid[b];
        int pos = offs[j] + atomicAdd(&cur[j], 1);
        list[pos] = b;
    }
}

// ---------- main: one wave processes 16 same-judge samples for all Q questions
__global__ __launch_bounds__(32) void calib_main(
    const float* __restrict__ x, const float* __restrict__ wc1,
    const float* __restrict__ wc2, const float* __restrict__ vc,
    const int* __restrict__ offs, const int* __restrict__ list,
    float* __restrict__ out)
{
    int j = blockIdx.y;
    int start = offs[j];
    int cnt = offs[j + 1] - start;
    int tile = blockIdx.x;
    if (tile * 16 >= cnt) return;

    int lane = threadIdx.x;
    int m = lane & 15;
    bool hi = lane >= 16;

    __shared__ float w2s[KP2 * WS2];   // Wc2^T tile: [i][h], row stride 72
    __shared__ float zb[16 * ZS];      // activation tile: [m][i]

    const float* wc2j = wc2 + j * KP2 * H2N;
    for (int t = lane; t < KP2 * H2N; t += 32)
        w2s[(t >> 6) * WS2 + (t & 63)] = wc2j[t];

    int s = tile * 16 + m;
    int b = (s < cnt) ? list[start + s] : -1;
    __syncthreads();

    for (int q = 0; q < Q_N; ++q) {
        // ----- build A fragments of xq = [1, x0..x4, 0, 0] (per-lane selects, no scratch)
        float x1v = 0.f, x2v = 0.f, x3v = 0.f, x4v = 0.f, x5v = 0.f;
        if (b >= 0) {
            const float* xp = x + (b * Q_N + q) * O_N;
            x1v = xp[0]; x2v = xp[1]; x3v = xp[2]; x4v = xp[3]; x5v = xp[4];
        }
        v2f a0, a1;
        a0.x = hi ? x2v : 1.0f;  a0.y = hi ? x3v : x1v;   // K = 0..3
        a1.x = hi ? 0.0f : x4v;  a1.y = hi ? 0.0f : x5v;  // K = 4..7

        // ----- layer 1: 16x64 = (16x8) x (8x64), 2 ksteps x 4 ntiles
        v8f acc[4];
#pragma unroll
        for (int nt = 0; nt < 4; ++nt) acc[nt] = (v8f)(0.0f);
        const float* w1j = wc1 + (j * Q_N + q) * KP1 * H1N;
#pragma unroll
        for (int ks = 0; ks < 2; ++ks) {
            int k0 = ks * 4 + (hi ? 2 : 0);
            v2f a = ks ? a1 : a0;
#pragma unroll
            for (int nt = 0; nt < 4; ++nt) {
                v2f bf;
                bf.x = w1j[k0 * H1N + nt * 16 + m];
                bf.y = w1j[(k0 + 1) * H1N + nt * 16 + m];
                acc[nt] = __builtin_amdgcn_wmma_f32_16x16x4_f32(
                    false, a, false, bf, (short)0, acc[nt], false, false);
            }
        }

        // ----- z1 = [1, relu(acc), 0-pad] into LDS (C layout -> [m][i] transpose)
#pragma unroll
        for (int nt = 0; nt < 4; ++nt) {
#pragma unroll
            for (int r = 0; r < 8; ++r) {
                int mm = (hi ? 8 : 0) + r;
                int n = nt * 16 + m;
                float v = acc[nt][r];
                zb[mm * ZS + 1 + n] = v > 0.f ? v : 0.f;
            }
        }
        if (lane < 16) {
            zb[lane * ZS + 0]  = 1.f;
            zb[lane * ZS + 65] = 0.f;
            zb[lane * ZS + 66] = 0.f;
            zb[lane * ZS + 67] = 0.f;
        }
        __syncthreads();

        // ----- layer 2: 16x64 = (16x68) x (68x64), 17 ksteps x 4 ntiles
        v8f acc2[4];
#pragma unroll
        for (int nt = 0; nt < 4; ++nt) acc2[nt] = (v8f)(0.0f);
#pragma unroll
        for (int ks = 0; ks < 17; ++ks) {
            int k0 = ks * 4 + (hi ? 2 : 0);
            v2f a;
            a.x = zb[m * ZS + k0];
            a.y = zb[m * ZS + k0 + 1];
#pragma unroll
            for (int nt = 0; nt < 4; ++nt) {
                v2f bf;
                bf.x = w2s[k0 * WS2 + nt * 16 + m];
                bf.y = w2s[(k0 + 1) * WS2 + nt * 16 + m];
                acc2[nt] = __builtin_amdgcn_wmma_f32_16x16x4_f32(
                    false, a, false, bf, (short)0, acc2[nt], false, false);
            }
        }
        __syncthreads();

        // ----- z2 = [1, relu(acc2), 0-pad] (bias/pad slots unchanged from z1)
#pragma unroll
        for (int nt = 0; nt < 4; ++nt) {
#pragma unroll
            for (int r = 0; r < 8; ++r) {
                int mm = (hi ? 8 : 0) + r;
                int n = nt * 16 + m;
                float v = acc2[nt][r];
                zb[mm * ZS + 1 + n] = v > 0.f ? v : 0.f;
            }
        }
        __syncthreads();

        // ----- layer 3 (65 -> 5) + softmax, lanes 0..15, one sample each
        if (lane < 16 && b >= 0) {
            const float* vj = vc + ((j * Q_N + q) * O_N) * (H2N + 1);
            float lg[O_N];
#pragma unroll
            for (int o = 0; o < O_N; ++o) {
                float ssum = 0.f;
                for (int i = 0; i < H2N + 1; ++i)
                    ssum += zb[lane * ZS + i] * vj[o * (H2N + 1) + i];
                lg[o] = ssum;
            }
            float mx = lg[0];
#pragma unroll
            for (int o = 1; o < O_N; ++o) mx = lg[o] > mx ? lg[o] : mx;
            float den = 0.f;
#pragma unroll
            for (int o = 0; o < O_N; ++o) { lg[o] = __expf(lg[o] - mx); den += lg[o]; }
            float inv = 1.0f / den;
            float* op = out + (b * Q_N + q) * O_N;
#pragma unroll
            for (int o = 0; o < O_N; ++o) op[o] = lg[o] * inv;
        }
        __syncthreads();
    }
}

extern "C" void kernel_launch(void* const* d_in, const int* in_sizes, int n_in,
                              void* d_out, int out_size, void* d_ws, size_t ws_size,
                              hipStream_t stream) {
    const float* x   = (const float*)d_in[0];
    const int*   jid = (const int*)d_in[1];
    const float* W1  = (const float*)d_in[2];
    const float* W1a = (const float*)d_in[3];
    const float* W2  = (const float*)d_in[4];
    const float* W2a = (const float*)d_in[5];
    const float* V   = (const float*)d_in[6];
    const float* Va  = (const float*)d_in[7];
    float* out = (float*)d_out;

    // workspace layout
    float* wc1 = (float*)d_ws;                               // 32*7*8*64   = 114688 f
    float* wc2 = wc1 + J_N * Q_N * KP1 * H1N;                // 32*68*64    = 139264 f
    float* vc  = wc2 + J_N * KP2 * H2N;                      // 32*7*5*65   = 72800 f
    int*   cnt  = (int*)(vc + J_N * Q_N * O_N * (H2N + 1));
    int*   cur  = cnt + J_N;
    int*   offs = cur + J_N;
    int*   list = offs + (J_N + 1);

    hipMemsetAsync(cnt, 0, 2 * J_N * sizeof(int), stream);

    combine_w1<<<(J_N * Q_N * KP1 * H1N + 255) / 256, 256, 0, stream>>>(W1, W1a, wc1);
    combine_w2<<<(J_N * KP2 * H2N + 255) / 256, 256, 0, stream>>>(W2, W2a, wc2);
    combine_vc<<<(J_N * Q_N * O_N * (H2N + 1) + 255) / 256, 256, 0, stream>>>(V, Va, vc);

    count_kernel<<<B_N / 256, 256, 0, stream>>>(jid, cnt);
    scan_kernel<<<1, 32, 0, stream>>>(cnt, offs);
    scatter_kernel<<<B_N / 256, 256, 0, stream>>>(jid, offs, cur, list);

    dim3 grid(B_N / 16, J_N);   // worst case: one judge owns all samples
    calib_main<<<grid, 32, 0, stream>>>(x, wc1, wc2, vc, offs, list, out);
}